// WaveCell_68169720922962
// MI455X (gfx1250) — compile-verified
//
#include <hip/hip_runtime.h>
#include <cstdint>

// CDNA5 / gfx1250 persistent wave-equation kernel, v4.1.
//  - 1 workgroup per batch; y1 plane LDS-resident (210x210 incl. zero halo
//    ring, ~172 KB of the 320 KB WGP LDS).
//  - Laplacian per 16x16 tile on the matrix pipe:  T'*Y + Y*T',
//    T' = tridiag(1,-2,1), via 8 back-to-back V_WMMA_F32_16X16X4_F32 with all
//    fragments preloaded; cross-tile halo rows/cols folded into C.
//  - y2 lives in 3 rotating global planes (L2-resident); ynew is transient.
//  - LDS refreshed each step by GLOBAL_LOAD_ASYNC_TO_LDS_B128 (ASYNCcnt).
//  - K coefficient loads kept per-iteration via "+v" anti-hoist asm (v4 used
//    "+s" on a divergent value -> illegal VGPR->SGPR copy; fixed).

typedef __attribute__((ext_vector_type(2))) float v2f;
typedef __attribute__((ext_vector_type(8))) float v8f;

#define NXg    200
#define NYg    200
#define Tt     1000
#define BATCHg 4
#define PADg   208            // 13 * 16
#define NTg    13             // tiles per dimension
#define TOTg   169            // 13*13 tiles
#define Wl     210            // plane row stride: 1 halo + 208 + 1 halo
#define PLANEg 44100          // Wl*Wl floats per plane
#define NWAVE  32             // waves per 1024-thread block
#define SLOTS  6              // ceil(169/32) tiles per wave
#define HHg    4.0401f        // H*H, H = 2.01

// ---------------- setup: per-cell coefficients K1,K2,K3 (padded, zero ring) --
__global__ void wave_setup_kernel(const float* __restrict__ rho,
                                  const float* __restrict__ bfield,
                                  float4* __restrict__ K4) {
  int idx = blockIdx.x * blockDim.x + threadIdx.x;
  if (idx >= PADg * PADg) return;
  int i = idx / PADg, j = idx % PADg;
  float4 k = make_float4(0.f, 0.f, 0.f, 0.f);
  if (i < NXg && j < NYg) {
    float b = bfield[i * NYg + j];
    float r = rho[i * NYg + j];
    float up = (i > 0)       ? rho[(i - 1) * NYg + j] : 0.f;
    float dn = (i < NXg - 1) ? rho[(i + 1) * NYg + j] : 0.f;
    float lf = (j > 0)       ? rho[i * NYg + j - 1]   : 0.f;
    float rt = (j < NYg - 1) ? rho[i * NYg + j + 1]   : 0.f;
    float lpf = 0.5f * r + 0.125f * (up + dn + lf + rt);
    const float BETA = 100.f, ETA = 0.5f;
    float t0   = tanhf(BETA * ETA);
    float proj = (t0 + tanhf(BETA * (lpf - ETA))) / (t0 + tanhf(BETA * (1.f - ETA)));
    float c  = 1.0f - 0.1f * proj;
    float c2 = c * c;
    float A1 = 1.f / (1.f + 0.5f * b);   // DT = 1
    k.x = A1 * 2.f;                      // K1 = A1*A2
    k.y = A1 * (1.f - 0.5f * b);         // K2 = A1*A3
    k.z = A1 * c2 / HHg;                 // K3 = A1*c2/H^2
  }
  K4[idx] = k;
}

// ---------------- zero the 12 rotating y-planes ------------------------------
__global__ void wave_zero_kernel(float* __restrict__ p, int n) {
  for (int i = blockIdx.x * blockDim.x + threadIdx.x; i < n;
       i += gridDim.x * blockDim.x)
    p[i] = 0.f;
}

// ---------------- main persistent time-stepping kernel ----------------------
__global__ void __launch_bounds__(1024)
wave_step_kernel(const float* __restrict__ x,
                 const float4* __restrict__ K4,
                 float* __restrict__ planes,
                 const int* __restrict__ psx, const int* __restrict__ psy,
                 const int* __restrict__ ppx, const int* __restrict__ ppy,
                 float* __restrict__ Istage) {
  extern __shared__ float ylds[];              // PLANEg floats, dynamic LDS
  const int tid  = threadIdx.x;
  const int wid  = tid >> 5;
  const int lane = tid & 31;
  const int lh   = lane >> 4;                  // lane half (wave32)
  const int ln   = lane & 15;
  const int bat  = blockIdx.x;

  float* pA = planes + ((size_t)bat * 3 + 0) * PLANEg;  // y_{t-1} archive
  float* pB = planes + ((size_t)bat * 3 + 1) * PLANEg;  // y_{t-2} (read)
  float* pC = planes + ((size_t)bat * 3 + 2) * PLANEg;  // y_t (write)

  for (int i = tid; i < PLANEg; i += 1024) ylds[i] = 0.f;

  // Constant tridiag(1,-2,1) fragments (WMMA A-layout, chunk k / vgpr v):
  // element = T'[M=ln][K = 4k+v+2*lh]; symmetric -> same regs serve as B.
  v2f aTv[4];
#pragma unroll
  for (int k = 0; k < 4; ++k)
#pragma unroll
    for (int v = 0; v < 2; ++v) {
      int d = ln - (4 * k + v + 2 * lh);
      aTv[k][v] = (d == 0) ? -2.f : ((d == 1 || d == -1) ? 1.f : 0.f);
    }

  // Lane-constant address addends (plane layout: cell (i,j) at (1+i)*Wl+1+j)
  const int dbY = 2 * lh * Wl + ln;                       // + (4k+v)*Wl
  const int daY = ln * Wl + 2 * lh;                       // + (4k+v)
  const int dD  = 8 * lh * Wl + ln;                       // + r*Wl
  const int dTB = (lh ? 16 * Wl : -Wl) + ln;
  const int dLR = 8 * lh * Wl + ((ln == 15) ? 16 : -1);   // + r*Wl
  const int dK  = 8 * lh * PADg + ln;                     // + r*PADg (K4 grid)
  const bool edgeCol = (ln == 0) || (ln == 15);

  const int sx = psx[0], sy = psy[0];
  const int px0 = ppx[0], px1 = ppx[1], px2 = ppx[2];
  const int py0 = ppy[0], py1 = ppy[1], py2 = ppy[2];

  int cellBase[SLOTS], kBase[SLOTS];
  unsigned validBits = 0u, wsrcBits = 0u, wprbBits = 0u;  // wave-uniform
  unsigned long long srcMask = 0ull, prbMask = 0ull;      // per-lane
  int prbIdx = -1;

#pragma unroll
  for (int s = 0; s < SLOTS; ++s) {
    int t = s * NWAVE + wid;
    bool vs = (t < TOTg);
    if (vs) validBits |= 1u << s;
    int tI = vs ? (t / NTg) : 0;
    int tJ = vs ? (t % NTg) : 0;
    int tM = tI * 16, tN = tJ * 16;
    cellBase[s] = (1 + tM) * Wl + 1 + tN;    // plane/LDS tile origin
    kBase[s]    = tM * PADg + tN;            // K4 tile origin
#pragma unroll
    for (int r = 0; r < 8; ++r) {
      int gM = tM + r + 8 * lh, gN = tN + ln;
      if (vs && gM == sx && gN == sy) srcMask |= 1ull << (s * 8 + r);
      if (vs) {
        int pj = -1;
        if (gM == px0 && gN == py0) pj = 0;
        if (gM == px1 && gN == py1) pj = 1;
        if (gM == px2 && gN == py2) pj = 2;
        if (pj >= 0) { prbMask |= 1ull << (s * 8 + r); prbIdx = pj; }
      }
    }
    if (__any((int)((srcMask >> (s * 8)) & 0xffull))) wsrcBits |= 1u << s;
    if (__any((int)((prbMask >> (s * 8)) & 0xffull))) wprbBits |= 1u << s;
  }

  float Iacc = 0.f;
  __syncthreads();                           // LDS (y1 = 0) visible

#pragma unroll 1
  for (int t = 0; t < Tt; ++t) {
    float xt = 0.f;
    if (wsrcBits) xt = x[bat * Tt + t];      // wave-uniform guard

    // ---------------- phase 1: compute y_t, store to pC --------------------
#pragma unroll
    for (int s = 0; s < SLOTS; ++s) {
      if (!((validBits >> s) & 1u)) continue;  // uniform: EXEC stays full
      const int cb = cellBase[s];

      // ---- preload ALL LDS operands for this tile (batched ds reads) ------
      v2f fb[4], fa[4];
#pragma unroll
      for (int k = 0; k < 4; ++k) {
        fb[k][0] = ylds[cb + dbY + (4 * k + 0) * Wl];  // Y rows   (B frags)
        fb[k][1] = ylds[cb + dbY + (4 * k + 1) * Wl];
        fa[k][0] = ylds[cb + daY + 4 * k + 0];         // Y cols   (A frags)
        fa[k][1] = ylds[cb + daY + 4 * k + 1];
      }
      float hTB = ylds[cb + dTB];            // lh0: top row, lh1: bottom row
      float hLR[8], y1d[8];
#pragma unroll
      for (int r = 0; r < 8; ++r) {
        hLR[r] = ylds[cb + dLR + r * Wl];    // ln0: left col, ln15: right col
        y1d[r] = ylds[cb + dD + r * Wl];     // own cells, C/D layout
      }

      // ---- accumulator C: cross-tile halo contributions -------------------
      v8f c;
#pragma unroll
      for (int r = 0; r < 8; ++r) c[r] = edgeCol ? hLR[r] : 0.f;
      c[0] += (lh == 0) ? hTB : 0.f;         // M=0  gets y[tM-1][n]
      c[7] += (lh == 1) ? hTB : 0.f;         // M=15 gets y[tM+16][n]

      // ---- lap = T'*Y + Y*T' : 8 back-to-back f32 WMMAs -------------------
#pragma unroll
      for (int k = 0; k < 4; ++k)            // T' * Y
        c = __builtin_amdgcn_wmma_f32_16x16x4_f32(false, aTv[k], false, fb[k],
                                                  (short)0, c, false, false);
#pragma unroll
      for (int k = 0; k < 4; ++k)            // Y * T'
        c = __builtin_amdgcn_wmma_f32_16x16x4_f32(false, fa[k], false, aTv[k],
                                                  (short)0, c, false, false);

      // ---- pointwise update; ynew fully transient -------------------------
      int kb = kBase[s];
      asm volatile("" : "+v"(kb));           // opaque: keep K loads in-loop
      const float*  y2p = pB + cb + dD;
      float*        ynp = pC + cb + dD;
      const float4* kp  = K4 + kb + dK;
#pragma unroll
      for (int r = 0; r < 8; ++r) {
        float4 kk = kp[r * PADg];            // b128, L2-resident
        float y2 = y2p[r * Wl];
        float yn = c[r] * kk.z;
        yn = fmaf(y1d[r], kk.x, yn);
        yn = fmaf(y2, -kk.y, yn);
        if ((wsrcBits >> s) & 1u)            // wave-uniform: 1 wave pays
          yn += ((srcMask >> (s * 8 + r)) & 1ull) ? xt : 0.f;
        if ((wprbBits >> s) & 1u) {          // wave-uniform: probes rare
          float v = ((prbMask >> (s * 8 + r)) & 1ull) ? yn : 0.f;
          Iacc = fmaf(v, v, Iacc);
        }
        ynp[r * Wl] = yn;
      }
    }

    // ---------------- phase 2: refresh LDS plane from pC (async DMA) -------
    asm volatile("s_wait_storecnt 0x0" ::: "memory");   // my stores at L2
    __syncthreads();                                    // everyone's stores done

    // copy floats [208, 208+43684) : rows 1..208 (+2 zero ring floats each
    // side so every b128 transfer is 16-byte aligned)
    for (int c4 = tid; c4 < 10921; c4 += 1024) {
      int off = 208 + c4 * 4;
      unsigned ldsOff = (unsigned)(off * 4);            // dyn-LDS starts at 0
      unsigned long long ga = (unsigned long long)(uintptr_t)(pC + off);
      asm volatile("global_load_async_to_lds_b128 %0, %1, off"
                   :: "v"(ldsOff), "v"(ga) : "memory");
    }
    asm volatile("s_wait_asynccnt 0x0" ::: "memory");
    __syncthreads();                                    // new y1 visible

    // rotate planes: (A,B,C) -> (C,A,B)
    float* tswap = pC; pC = pB; pB = pA; pA = tswap;
  }

  if (prbIdx >= 0) Istage[bat * 3 + prbIdx] = Iacc;     // unique writer
}

// ---------------- final normalization (12 outputs) --------------------------
__global__ void wave_norm_kernel(const float* __restrict__ Istage,
                                 float* __restrict__ out) {
  int bb = threadIdx.x;
  if (bb < BATCHg) {
    float i0 = Istage[bb * 3 + 0];
    float i1 = Istage[bb * 3 + 1];
    float i2 = Istage[bb * 3 + 2];
    float s = i0 + i1 + i2;
    out[bb * 3 + 0] = i0 / s;
    out[bb * 3 + 1] = i1 / s;
    out[bb * 3 + 2] = i2 / s;
  }
}

extern "C" void kernel_launch(void* const* d_in, const int* in_sizes, int n_in,
                              void* d_out, int out_size, void* d_ws, size_t ws_size,
                              hipStream_t stream) {
  const float* x   = (const float*)d_in[0];   // (4,1000) f32
  const float* rho = (const float*)d_in[1];   // (200,200) f32
  const float* bfd = (const float*)d_in[2];   // (200,200) f32
  const int* sx = (const int*)d_in[3];
  const int* sy = (const int*)d_in[4];
  const int* px = (const int*)d_in[5];        // (3,) i32
  const int* py = (const int*)d_in[6];        // (3,) i32

  // ws layout: K4 (692,224 B) | 12 planes (2,116,800 B) | Istage (48 B)
  float4* K4     = (float4*)d_ws;
  float*  planes = (float*)((char*)d_ws + (size_t)PADg * PADg * sizeof(float4));
  float*  Istage = planes + (size_t)BATCHg * 3 * PLANEg;

  wave_setup_kernel<<<(PADg * PADg + 255) / 256, 256, 0, stream>>>(rho, bfd, K4);
  wave_zero_kernel<<<512, 256, 0, stream>>>(planes, BATCHg * 3 * PLANEg);
  wave_step_kernel<<<BATCHg, 1024, PLANEg * sizeof(float), stream>>>(
      x, K4, planes, sx, sy, px, py, Istage);
  wave_norm_kernel<<<1, 64, 0, stream>>>(Istage, (float*)d_out);
}